// GINModel_22892175688472
// MI455X (gfx1250) — compile-verified
//
#include <hip/hip_runtime.h>

#define N_NODES 100000
#define C 128
#define TILE_ROWS 64        // rows per block (4 waves x 16 rows)
#define LDS_STRIDE (C + 4)  // pad so A-fragment reads hit distinct banks

typedef __attribute__((ext_vector_type(2))) float v2f;
typedef __attribute__((ext_vector_type(8))) float v8f;

// ---------------------------------------------------------------------------
// agg = src (initializes the poisoned workspace deterministically)
// ---------------------------------------------------------------------------
__global__ void copy_f4(const float4* __restrict__ in, float4* __restrict__ out, int n4) {
  int i = blockIdx.x * blockDim.x + threadIdx.x;
  if (i < n4) out[i] = in[i];
}

// ---------------------------------------------------------------------------
// agg[dst] += feat[src]  — one wave32 per edge, float4 per lane (512B rows)
// ---------------------------------------------------------------------------
__global__ void scatter_add(const float* __restrict__ feat, float* __restrict__ agg,
                            const int* __restrict__ srcIdx, const int* __restrict__ dstIdx,
                            int nEdges) {
  int t = blockIdx.x * blockDim.x + threadIdx.x;
  int e = t >> 5;
  if (e >= nEdges) return;
  int lane = t & 31;
  int s = srcIdx[e];
  int d = dstIdx[e];
  float4 v = ((const float4*)(feat + (size_t)s * C))[lane];
  float* dp = agg + (size_t)d * C + lane * 4;
  unsafeAtomicAdd(dp + 0, v.x);
  unsafeAtomicAdd(dp + 1, v.y);
  unsafeAtomicAdd(dp + 2, v.z);
  unsafeAtomicAdd(dp + 3, v.w);
}

// ---------------------------------------------------------------------------
// Wave-level helper: 16x128 (LDS rows) @ 128x128 (global W) via f32 WMMA.
// A fragment (16x4 f32):  lane<16 -> M=lane, K=kk+{0,1}; lane>=16 -> K=kk+{2,3}
// B fragment (4x16 f32):  lane<16 -> N=lane, K=kk+{0,1}; lane>=16 -> K=kk+{2,3}
// ---------------------------------------------------------------------------
__device__ __forceinline__ void wave_gemm_16x128(
    const float (*hs)[LDS_STRIDE], int wr, int half, int lcol,
    const float* __restrict__ W, v8f accs[8]) {
  for (int kk = 0; kk < C; kk += 4) {
    const int k0 = kk + half * 2;
    v2f a;
    a.x = hs[wr + lcol][k0];
    a.y = hs[wr + lcol][k0 + 1];
    const float* wp = W + (size_t)k0 * C;
#pragma unroll
    for (int nn = 0; nn < 8; ++nn) {
      v2f b;
      b.x = wp[nn * 16 + lcol];        // row k0,   col nn*16+lcol
      b.y = wp[C + nn * 16 + lcol];    // row k0+1, col nn*16+lcol
      accs[nn] = __builtin_amdgcn_wmma_f32_16x16x4_f32(
          false, a, false, b, (short)0, accs[nn], false, false);
    }
  }
}

// ---------------------------------------------------------------------------
// Fused GIN MLP:  Hout = [relu]( relu(Hin@Wa + ba) @ Wb + bb )
// ---------------------------------------------------------------------------
__global__ __launch_bounds__(128)
void gin_mlp(const float* __restrict__ Hin,
             const float* __restrict__ Wa, const float* __restrict__ ba,
             const float* __restrict__ Wb, const float* __restrict__ bb,
             float* __restrict__ Hout, int reluOut) {
  __shared__ float hs[TILE_ROWS][LDS_STRIDE];
  const int tid = threadIdx.x;
  const int wave = tid >> 5;
  const int lane = tid & 31;
  const int rowBase = blockIdx.x * TILE_ROWS;

  // stage 64x128 tile (zero-fill OOB rows)
  for (int it = 0; it < 16; ++it) {
    int idx = it * 128 + tid;
    int r = idx >> 5, c4 = idx & 31;
    int grow = rowBase + r;
    float4 v = make_float4(0.f, 0.f, 0.f, 0.f);
    if (grow < N_NODES) v = ((const float4*)(Hin + (size_t)grow * C))[c4];
    float* p = &hs[r][c4 * 4];
    p[0] = v.x; p[1] = v.y; p[2] = v.z; p[3] = v.w;
  }
  __syncthreads();

  const int wr = wave * 16;
  const int half = lane >> 4;
  const int lcol = lane & 15;
  const int mbase = half * 8;   // D layout: VGPR r holds M=r (lanes 0-15) / M=8+r (16-31)
  const v8f vzero = {0, 0, 0, 0, 0, 0, 0, 0};

  v8f accs[8];

  // ---- GEMM1: t = relu(h @ Wa + ba); write back in-place to this wave's rows
#pragma unroll
  for (int nn = 0; nn < 8; ++nn) accs[nn] = vzero;
  wave_gemm_16x128(hs, wr, half, lcol, Wa, accs);
#pragma unroll
  for (int nn = 0; nn < 8; ++nn) {
    float bias = ba[nn * 16 + lcol];
#pragma unroll
    for (int r = 0; r < 8; ++r) {
      float v = accs[nn][r] + bias;
      v = v > 0.f ? v : 0.f;
      hs[wr + mbase + r][nn * 16 + lcol] = v;  // wave-private rows, no barrier needed
    }
  }

  // ---- GEMM2: out = [relu](t @ Wb + bb)
#pragma unroll
  for (int nn = 0; nn < 8; ++nn) accs[nn] = vzero;
  wave_gemm_16x128(hs, wr, half, lcol, Wb, accs);
#pragma unroll
  for (int nn = 0; nn < 8; ++nn) {
    float bias = bb[nn * 16 + lcol];
#pragma unroll
    for (int r = 0; r < 8; ++r) {
      float v = accs[nn][r] + bias;
      if (reluOut) v = v > 0.f ? v : 0.f;
      int grow = rowBase + wr + mbase + r;
      if (grow < N_NODES) Hout[(size_t)grow * C + nn * 16 + lcol] = v;
    }
  }
}

// ---------------------------------------------------------------------------
// Layer 3: out = relu(Hin@W3a + b3a) @ w3b(128x1) + b3b  (per-row dot product)
// ---------------------------------------------------------------------------
__global__ __launch_bounds__(128)
void gin_mlp3(const float* __restrict__ Hin,
              const float* __restrict__ Wa, const float* __restrict__ ba,
              const float* __restrict__ wb, const float* __restrict__ bb,
              float* __restrict__ out) {
  __shared__ float hs[TILE_ROWS][LDS_STRIDE];
  const int tid = threadIdx.x;
  const int wave = tid >> 5;
  const int lane = tid & 31;
  const int rowBase = blockIdx.x * TILE_ROWS;

  for (int it = 0; it < 16; ++it) {
    int idx = it * 128 + tid;
    int r = idx >> 5, c4 = idx & 31;
    int grow = rowBase + r;
    float4 v = make_float4(0.f, 0.f, 0.f, 0.f);
    if (grow < N_NODES) v = ((const float4*)(Hin + (size_t)grow * C))[c4];
    float* p = &hs[r][c4 * 4];
    p[0] = v.x; p[1] = v.y; p[2] = v.z; p[3] = v.w;
  }
  __syncthreads();

  const int wr = wave * 16;
  const int half = lane >> 4;
  const int lcol = lane & 15;
  const int mbase = half * 8;
  const v8f vzero = {0, 0, 0, 0, 0, 0, 0, 0};

  v8f accs[8];
#pragma unroll
  for (int nn = 0; nn < 8; ++nn) accs[nn] = vzero;
  wave_gemm_16x128(hs, wr, half, lcol, Wa, accs);
#pragma unroll
  for (int nn = 0; nn < 8; ++nn) {
    float bias = ba[nn * 16 + lcol];
#pragma unroll
    for (int r = 0; r < 8; ++r) {
      float v = accs[nn][r] + bias;
      v = v > 0.f ? v : 0.f;
      hs[wr + mbase + r][nn * 16 + lcol] = v;
    }
  }

  // per-row dot with the 128x1 output weight
  float4 wv = ((const float4*)wb)[lane];
  float bout = bb[0];
  for (int r = 0; r < 16; ++r) {
    const float* tp = &hs[wr + r][lane * 4];
    float p = tp[0] * wv.x + tp[1] * wv.y + tp[2] * wv.z + tp[3] * wv.w;
#pragma unroll
    for (int off = 16; off > 0; off >>= 1) p += __shfl_down(p, off, 32);
    if (lane == 0) {
      int grow = rowBase + wr + r;
      if (grow < N_NODES) out[grow] = p + bout;
    }
  }
}

// ---------------------------------------------------------------------------
extern "C" void kernel_launch(void* const* d_in, const int* in_sizes, int n_in,
                              void* d_out, int out_size, void* d_ws, size_t ws_size,
                              hipStream_t stream) {
  const float* x   = (const float*)d_in[0];
  const int*   ei  = (const int*)d_in[1];
  const float* W1a = (const float*)d_in[2];  const float* b1a = (const float*)d_in[3];
  const float* W1b = (const float*)d_in[4];  const float* b1b = (const float*)d_in[5];
  const float* W2a = (const float*)d_in[6];  const float* b2a = (const float*)d_in[7];
  const float* W2b = (const float*)d_in[8];  const float* b2b = (const float*)d_in[9];
  const float* W3a = (const float*)d_in[10]; const float* b3a = (const float*)d_in[11];
  const float* W3b = (const float*)d_in[12]; const float* b3b = (const float*)d_in[13];
  float* out = (float*)d_out;

  const int nEdges = in_sizes[1] / 2;       // edge_index is (2, E) flattened
  const int* src = ei;
  const int* dst = ei + nEdges;

  float* agg  = (float*)d_ws;                     // 100000*128 f32 = 51.2 MB
  float* hbuf = agg + (size_t)N_NODES * C;        // second 51.2 MB

  const int n4 = N_NODES * C / 4;
  dim3 cpb(256), cpg((n4 + 255) / 256);
  dim3 scb(256), scg((unsigned)((nEdges * 32 + 255) / 256));
  const int mlpBlocks = (N_NODES + TILE_ROWS - 1) / TILE_ROWS;

  // ---- layer 1: agg = x + scatter(x);  hbuf = relu(MLP1(agg))
  copy_f4<<<cpg, cpb, 0, stream>>>((const float4*)x, (float4*)agg, n4);
  scatter_add<<<scg, scb, 0, stream>>>(x, agg, src, dst, nEdges);
  gin_mlp<<<mlpBlocks, 128, 0, stream>>>(agg, W1a, b1a, W1b, b1b, hbuf, 1);

  // ---- layer 2
  copy_f4<<<cpg, cpb, 0, stream>>>((const float4*)hbuf, (float4*)agg, n4);
  scatter_add<<<scg, scb, 0, stream>>>(hbuf, agg, src, dst, nEdges);
  gin_mlp<<<mlpBlocks, 128, 0, stream>>>(agg, W2a, b2a, W2b, b2b, hbuf, 1);

  // ---- layer 3 (output dim 1, no outer relu)
  copy_f4<<<cpg, cpb, 0, stream>>>((const float4*)hbuf, (float4*)agg, n4);
  scatter_add<<<scg, scb, 0, stream>>>(hbuf, agg, src, dst, nEdges);
  gin_mlp3<<<mlpBlocks, 128, 0, stream>>>(agg, W3a, b3a, W3b, b3b, out);
}